// SVD_22565758174141
// MI455X (gfx1250) — compile-verified
//
#include <hip/hip_runtime.h>

// Weighted Kabsch / Procrustes for MI455X (gfx1250, wave32).
// Phase 1: per-batch augmented 4xN * Nx4 reduction on the matrix pipe via
//          V_WMMA_F32_16X16X4_F32. 4 batches packed as diagonal 4x4 blocks of
//          one 16x16 tile; N=1024 split over 8 waves (128 pts each) for
//          bandwidth-saturating concurrency (8192 waves total).
// Phase 2: LDS reduction of split partials + 3x3 Jacobi SVD, 1 thread/batch.

typedef float v2f __attribute__((ext_vector_type(2)));
typedef float v4f __attribute__((ext_vector_type(4)));
typedef float v8f __attribute__((ext_vector_type(8)));

#define B_TOT 4096
#define NPTS  1024
#define SPLIT 8                    // waves per batch-group
#define PTS_PER_WAVE (NPTS / SPLIT)  // 128
#define ITERS (PTS_PER_WAVE / 8)     // 16 (8 points per iteration)

__device__ __forceinline__ void wmma_pair(v4f w4, v4f t4, v4f s4, bool isCoord,
                                          v8f& c0, v8f& c1) {
  v4f wt = w4 * t4;
  v4f a4 = isCoord ? wt : w4;                     // A rows: w*t_c | w
  v4f b4 = isCoord ? s4 : (v4f){1.f, 1.f, 1.f, 1.f};  // B cols: s_c | 1
  v2f a0 = {a4.x, a4.y}, a1 = {a4.z, a4.w};
  v2f b0 = {b4.x, b4.y}, b1 = {b4.z, b4.w};
  c0 = __builtin_amdgcn_wmma_f32_16x16x4_f32(false, a0, false, b0, (short)0, c0,
                                             false, false);
  c1 = __builtin_amdgcn_wmma_f32_16x16x4_f32(false, a1, false, b1, (short)0, c1,
                                             false, false);
}

__device__ __forceinline__ void jrot(float A[3][3], float V[3][3], int p, int q) {
  float apq = A[p][q];
  if (fabsf(apq) < 1e-24f) return;
  float tau = (A[q][q] - A[p][p]) / (2.0f * apq);
  float t   = copysignf(1.0f, tau) / (fabsf(tau) + sqrtf(1.0f + tau * tau));
  float c   = rsqrtf(1.0f + t * t);
  float s   = t * c;
#pragma unroll
  for (int k = 0; k < 3; ++k) {
    float akp = A[k][p], akq = A[k][q];
    A[k][p] = c * akp - s * akq;
    A[k][q] = s * akp + c * akq;
  }
#pragma unroll
  for (int k = 0; k < 3; ++k) {
    float apk = A[p][k], aqk = A[q][k];
    A[p][k] = c * apk - s * aqk;
    A[q][k] = s * apk + c * aqk;
  }
#pragma unroll
  for (int k = 0; k < 3; ++k) {
    float vkp = V[k][p], vkq = V[k][q];
    V[k][p] = c * vkp - s * vkq;
    V[k][q] = s * vkp + c * vkq;
  }
}

__device__ __forceinline__ void cross3(const float a[3], const float b[3], float o[3]) {
  o[0] = a[1] * b[2] - a[2] * b[1];
  o[1] = a[2] * b[0] - a[0] * b[2];
  o[2] = a[0] * b[1] - a[1] * b[0];
}

__global__ __launch_bounds__(256) void procrustes_wmma_kernel(
    const float* __restrict__ src, const float* __restrict__ tgt,
    const float* __restrict__ wts, float* __restrict__ out) {
  // [split-wave][batch-in-group][col][row] partial 4x4 augmented sums
  __shared__ float lds[SPLIT][4][4][4];

  const int tid  = threadIdx.x;
  const int lane = tid & 31;
  const int wv   = tid >> 5;           // split index 0..7
  const int bblk = blockIdx.x * 4;     // this block's 4-batch group

  // A row m == B col n == lane&15 ; decompose into (batch-in-group, component)
  const int m   = lane & 15;
  const int kh  = lane >> 4;           // K-half this lane feeds
  const int abb = m >> 2;              // batch within group
  const int ac  = m & 3;               // 0..2 = coord, 3 = augmentation
  const int b   = bblk + abb;
  const bool isCoord = (ac < 3);
  const int  crow    = isCoord ? ac : 0;

  const float* tptr = tgt + (size_t)(b * 3 + crow) * NPTS;
  const float* sptr = src + (size_t)(b * 3 + crow) * NPTS;
  const float* wptr = wts + (size_t)b * NPTS;

  v8f c0 = {0.f, 0.f, 0.f, 0.f, 0.f, 0.f, 0.f, 0.f};
  v8f c1 = {0.f, 0.f, 0.f, 0.f, 0.f, 0.f, 0.f, 0.f};

  // K-slot -> point mapping per step: K{0,1}->q,q+1 (kh=0), K{2,3}->q+4,q+5
  // (kh=1); one b128 per stream feeds two WMMAs with contiguous lane data.
  int q = wv * PTS_PER_WAVE + 4 * kh;
  v4f wc = *(const v4f*)(wptr + q);
  v4f tc = *(const v4f*)(tptr + q);
  v4f sc = *(const v4f*)(sptr + q);
  for (int it = 0; it < ITERS - 1; ++it) {
    const int qn = q + 8;
    v4f wn = *(const v4f*)(wptr + qn);   // next stage issued before consume
    v4f tn = *(const v4f*)(tptr + qn);
    v4f sn = *(const v4f*)(sptr + qn);
    wmma_pair(wc, tc, sc, isCoord, c0, c1);
    wc = wn; tc = tn; sc = sn; q = qn;
  }
  wmma_pair(wc, tc, sc, isCoord, c0, c1);

  v8f cc = c0 + c1;
  float carr[8];
#pragma unroll
  for (int i = 0; i < 8; ++i) carr[i] = cc[i];

  // C layout: VGPR v, lanes 0-15 -> row v, lanes 16-31 -> row v+8; col = lane&15.
  // Diagonal 4x4 block of batch bn lives in rows 4bn..4bn+3: owned by kh == bn>>1.
  const int n  = lane & 15;
  const int bn = n >> 2;
  if ((bn >> 1) == kh) {
    const bool hi = (bn & 1);
#pragma unroll
    for (int r = 0; r < 4; ++r)
      lds[wv][bn][n & 3][r] = hi ? carr[4 + r] : carr[r];
  }
  __syncthreads();

  if (tid < 4) {
    const int gb = blockIdx.x * 4 + tid;

    float M[4][4];
#pragma unroll
    for (int r = 0; r < 4; ++r)
#pragma unroll
      for (int cI = 0; cI < 4; ++cI) {
        float acc = 0.f;
#pragma unroll
        for (int s = 0; s < SPLIT; ++s) acc += lds[s][tid][cI][r];
        M[r][cI] = acc;
      }

    const float wsum = M[3][3];
    const float invw = 1.0f / wsum;
    float ct[3], cs[3], H[3][3];
#pragma unroll
    for (int i = 0; i < 3; ++i) {
      ct[i] = M[i][3] * invw;  // tgt centroid
      cs[i] = M[3][i] * invw;  // src centroid
    }
#pragma unroll
    for (int i = 0; i < 3; ++i)
#pragma unroll
      for (int j = 0; j < 3; ++j)
        H[i][j] = (M[i][j] - M[i][3] * M[3][j] * invw) * invw;

    // S = H^T H, Jacobi eigendecomposition -> V, eigenvalues
    float S[3][3], V[3][3] = {{1.f, 0.f, 0.f}, {0.f, 1.f, 0.f}, {0.f, 0.f, 1.f}};
#pragma unroll
    for (int i = 0; i < 3; ++i)
#pragma unroll
      for (int j = 0; j < 3; ++j)
        S[i][j] = H[0][i] * H[0][j] + H[1][i] * H[1][j] + H[2][i] * H[2][j];
    for (int sweep = 0; sweep < 6; ++sweep) {
      jrot(S, V, 0, 1);
      jrot(S, V, 0, 2);
      jrot(S, V, 1, 2);
    }
    float l0 = S[0][0], l1 = S[1][1], l2 = S[2][2];
#define SWAPCOL(a, b)                            \
  {                                              \
    for (int k = 0; k < 3; ++k) {                \
      float tmp = V[k][a];                       \
      V[k][a] = V[k][b];                         \
      V[k][b] = tmp;                             \
    }                                            \
  }
    if (l0 < l1) { float t = l0; l0 = l1; l1 = t; SWAPCOL(0, 1); }
    if (l0 < l2) { float t = l0; l0 = l2; l2 = t; SWAPCOL(0, 2); }
    if (l1 < l2) { float t = l1; l1 = l2; l2 = t; SWAPCOL(1, 2); }
#undef SWAPCOL

    float v0[3] = {V[0][0], V[1][0], V[2][0]};
    float v1[3] = {V[0][1], V[1][1], V[2][1]};
    float v2c[3];
    cross3(v0, v1, v2c);  // force det(V) = +1

    float u0[3], u1[3], u2[3];
#pragma unroll
    for (int i = 0; i < 3; ++i)
      u0[i] = H[i][0] * v0[0] + H[i][1] * v0[1] + H[i][2] * v0[2];
    float n0 = sqrtf(u0[0] * u0[0] + u0[1] * u0[1] + u0[2] * u0[2]);
    if (n0 > 1e-12f) {
      float r = 1.0f / n0;
      u0[0] *= r; u0[1] *= r; u0[2] *= r;
    } else {
      u0[0] = 1.f; u0[1] = 0.f; u0[2] = 0.f;
    }
#pragma unroll
    for (int i = 0; i < 3; ++i)
      u1[i] = H[i][0] * v1[0] + H[i][1] * v1[1] + H[i][2] * v1[2];
    float d01 = u0[0] * u1[0] + u0[1] * u1[1] + u0[2] * u1[2];
    u1[0] -= d01 * u0[0]; u1[1] -= d01 * u0[1]; u1[2] -= d01 * u0[2];
    float n1 = sqrtf(u1[0] * u1[0] + u1[1] * u1[1] + u1[2] * u1[2]);
    if (n1 > 1e-12f) {
      float r = 1.0f / n1;
      u1[0] *= r; u1[1] *= r; u1[2] *= r;
    } else {
      float ax[3] = {0.f, 0.f, 0.f};
      ax[fabsf(u0[0]) < 0.9f ? 0 : 1] = 1.0f;
      float d = ax[0] * u0[0] + ax[1] * u0[1] + ax[2] * u0[2];
      u1[0] = ax[0] - d * u0[0]; u1[1] = ax[1] - d * u0[1]; u1[2] = ax[2] - d * u0[2];
      float rn = rsqrtf(u1[0] * u1[0] + u1[1] * u1[1] + u1[2] * u1[2]);
      u1[0] *= rn; u1[1] *= rn; u1[2] *= rn;
    }
    cross3(u0, u1, u2);  // det(U) = +1 -> det-sign correction implicit

    float R[3][3];
#pragma unroll
    for (int i = 0; i < 3; ++i)
#pragma unroll
      for (int j = 0; j < 3; ++j)
        R[i][j] = u0[i] * v0[j] + u1[i] * v1[j] + u2[i] * v2c[j];

    float t1[3], t2[3];
#pragma unroll
    for (int i = 0; i < 3; ++i)
      t1[i] = cs[i] - (R[0][i] * ct[0] + R[1][i] * ct[1] + R[2][i] * ct[2]);
#pragma unroll
    for (int i = 0; i < 3; ++i)
      t2[i] = -(R[i][0] * t1[0] + R[i][1] * t1[1] + R[i][2] * t1[2]);

    // Outputs: T (B,4,4) | R (B,3,3) | t (B,3,1), flat in return order.
    float* T    = out + (size_t)gb * 16;
    float* Rout = out + (size_t)B_TOT * 16 + (size_t)gb * 9;
    float* tout = out + (size_t)B_TOT * 25 + (size_t)gb * 3;
#pragma unroll
    for (int i = 0; i < 3; ++i) {
#pragma unroll
      for (int j = 0; j < 3; ++j) {
        T[i * 4 + j]    = R[i][j];
        Rout[i * 3 + j] = R[i][j];
      }
      T[i * 4 + 3] = t2[i];
      tout[i]      = t2[i];
    }
    T[12] = 0.f; T[13] = 0.f; T[14] = 0.f; T[15] = 1.f;
  }
}

extern "C" void kernel_launch(void* const* d_in, const int* in_sizes, int n_in,
                              void* d_out, int out_size, void* d_ws, size_t ws_size,
                              hipStream_t stream) {
  (void)in_sizes; (void)n_in; (void)out_size; (void)d_ws; (void)ws_size;
  const float* src = (const float*)d_in[0];
  const float* tgt = (const float*)d_in[1];
  const float* wts = (const float*)d_in[2];
  float* out = (float*)d_out;
  // One 4-batch group per 256-thread block; N split across its 8 waves.
  // 1024 blocks -> 8192 waves streaming 117 MB (~5 us floor at 23.3 TB/s).
  procrustes_wmma_kernel<<<dim3(B_TOT / 4), dim3(256), 0, stream>>>(src, tgt, wts, out);
}